// GodContinuousAreaModel_5729486373546
// MI455X (gfx1250) — compile-verified
//
#include <hip/hip_runtime.h>
#include <math.h>

// GodContinuousAreaModel for MI455X (gfx1250, wave32, WMMA).
// Fuses: gated/scaled synapse GEMM (32x8192 @ 8192x8192^T) + receptor GEMM
// (32x8192 from x @ Wr^T) + biases + tanh, using v_wmma_f32_16x16x4_f32
// (full f32 precision; the kernel is HBM-bound on the 256MB synapse matrix,
// ~11us at 23.3 TB/s, so f32 matrix ops are the right precision choice).

typedef __attribute__((ext_vector_type(2))) float v2f;
typedef __attribute__((ext_vector_type(8))) float v8f;

#define B_SZ    32
#define D_INK   1024
#define N_AREAS 4
#define N_PER   2048
#define N_TOT   8192
#define OUTD    11
#define KBLK    512
#define LDS_PITCH (KBLK + 2)   // pad: row stride 514 dwords -> banks spread, keeps 8B align

// ---------------- gate[a] = (mean |Z0[:,a,:]| > 0.05) ? 1 : 0 ----------------
__global__ void gate_kernel(const float* __restrict__ Z0, float* __restrict__ gate) {
    __shared__ float red[256];
    const int a = blockIdx.x;
    const int tid = threadIdx.x;
    float s = 0.0f;
    for (int i = tid; i < B_SZ * N_PER; i += 256) {
        int b = i / N_PER, m = i % N_PER;
        s += fabsf(Z0[(size_t)b * N_TOT + a * N_PER + m]);
    }
    red[tid] = s;
    __syncthreads();
    for (int o = 128; o > 0; o >>= 1) {
        if (tid < o) red[tid] += red[tid + o];
        __syncthreads();
    }
    if (tid == 0)
        gate[a] = (red[0] * (1.0f / (float)(B_SZ * N_PER)) > 0.05f) ? 1.0f : 0.0f;
}

// ---------------- fused main GEMM using v_wmma_f32_16x16x4_f32 ----------------
// acc[b, j] += sum_k W[j,k] * (Act[b,k] * fac(karea))
// A-frag (16x4 f32): lane L holds Act[M=L&15][K = 2*(L>>4) + {0,1}]   (LDS, ds_load_b64)
// B-frag (4x16 f32): lane L holds W[j = jtile + (L&15)][K = 2*(L>>4) + {0,1}] (global b64)
template <bool SCALED>
__device__ __forceinline__ void gemm_accum(v8f& acc0, v8f& acc1,
                                           const float* __restrict__ W,
                                           const float* __restrict__ Act,
                                           int K, int jtile, int jarea,
                                           const float* __restrict__ gate,
                                           float* __restrict__ Zs) {
    const int tid  = threadIdx.x;
    const int lane = tid & 31;
    const int row  = lane & 15;
    const int h    = lane >> 4;

    for (int kb = 0; kb < K; kb += KBLK) {
        float fac = 1.0f;
        if (SCALED) {
            int karea = kb >> 11;                       // KBLK divides N_PER, so uniform
            fac = gate[karea] * ((karea == jarea) ? 0.8f : 0.1f);
        }
        __syncthreads();                                // WAR: prior readers done
        // Stage 32 x KBLK activations into LDS, pre-scaled (coalesced float4 loads).
        for (int i = tid * 4; i < B_SZ * KBLK; i += 128 * 4) {
            int b = i / KBLK, kk = i % KBLK;
            const float4 v = *(const float4*)(Act + (size_t)b * K + kb + kk);
            float* dst = Zs + b * LDS_PITCH + kk;
            dst[0] = v.x * fac; dst[1] = v.y * fac;
            dst[2] = v.z * fac; dst[3] = v.w * fac;
        }
        __syncthreads();

        const float* wrow  = W  + (size_t)(jtile + row) * K + kb + 2 * h;
        if (kb + KBLK < K) __builtin_prefetch(wrow + KBLK, 0, 1);  // global_prefetch_b8
        const float* arow0 = Zs + (row)      * LDS_PITCH + 2 * h;
        const float* arow1 = Zs + (row + 16) * LDS_PITCH + 2 * h;

#pragma unroll 4
        for (int k = 0; k < KBLK; k += 4) {
            v2f bf = *(const v2f*)(wrow  + k);   // W[j, k..k+1] for this lane
            v2f a0 = *(const v2f*)(arow0 + k);   // Act[b=row,    k..k+1]
            v2f a1 = *(const v2f*)(arow1 + k);   // Act[b=row+16, k..k+1]
            acc0 = __builtin_amdgcn_wmma_f32_16x16x4_f32(
                       false, a0, false, bf, (short)0, acc0, false, false);
            acc1 = __builtin_amdgcn_wmma_f32_16x16x4_f32(
                       false, a1, false, bf, (short)0, acc1, false, false);
        }
    }
}

__global__ void __launch_bounds__(128)
fused_kernel(const float* __restrict__ x,  const float* __restrict__ Z0,
             const float* __restrict__ Wr, const float* __restrict__ br,
             const float* __restrict__ Ws, const float* __restrict__ bs,
             const float* __restrict__ gate, float* __restrict__ Znew) {
    __shared__ float Zs[B_SZ * LDS_PITCH];
    const int wave  = threadIdx.x >> 5;          // 4 waves per block
    const int lane  = threadIdx.x & 31;
    const int jtile = blockIdx.x * 64 + wave * 16;
    const int jarea = jtile >> 11;               // block's 64 cols lie in one area

    v8f acc0 = {};   // batch rows 0..15
    v8f acc1 = {};   // batch rows 16..31

    // Znext: gated/scaled synapse GEMM over the full 256MB weight matrix.
    gemm_accum<true >(acc0, acc1, Ws, Z0, N_TOT, jtile, jarea, gate, Zs);
    // stim:  receptor GEMM accumulated into the same tile.
    gemm_accum<false>(acc0, acc1, Wr, x,  D_INK, jtile, jarea, gate, Zs);

    // Epilogue: + rec_b + gate*syn_b, tanh, store.
    // C/D layout: VGPR g -> M = g + 8*(lane>>4), N = lane&15.
    const int n = lane & 15, h = lane >> 4;
    const int j = jtile + n;
    const float bias = br[j] + gate[jarea] * bs[j];
#pragma unroll
    for (int g = 0; g < 8; ++g) {
        const int b = g + 8 * h;
        Znew[(size_t)(b)      * N_TOT + j] = tanhf(acc0[g] + bias);
        Znew[(size_t)(b + 16) * N_TOT + j] = tanhf(acc1[g] + bias);
    }
}

// ---------------- output lobe: raw = Znew @ Wo^T + bo; split logits/sigmoid ----
__global__ void out_kernel(const float* __restrict__ Znew, const float* __restrict__ Wo,
                           const float* __restrict__ bo, float* __restrict__ out) {
    __shared__ float red[256];
    const int b = blockIdx.x / OUTD;
    const int o = blockIdx.x % OUTD;
    const int tid = threadIdx.x;
    const float* z = Znew + (size_t)b * N_TOT;
    const float* w = Wo   + (size_t)o * N_TOT;
    float s = 0.0f;
    for (int j = tid; j < N_TOT; j += 256) s += z[j] * w[j];
    red[tid] = s;
    __syncthreads();
    for (int t = 128; t > 0; t >>= 1) {
        if (tid < t) red[tid] += red[tid + t];
        __syncthreads();
    }
    if (tid == 0) {
        float v = red[0] + bo[o];
        if (o < 10) out[b * 10 + o] = v;                       // logits (32,10)
        else        out[320 + b]    = 1.0f / (1.0f + expf(-v)); // out_gate (32,)
    }
}

// ---------------- launcher ----------------
extern "C" void kernel_launch(void* const* d_in, const int* in_sizes, int n_in,
                              void* d_out, int out_size, void* d_ws, size_t ws_size,
                              hipStream_t stream) {
    const float* x  = (const float*)d_in[0];
    const float* Z0 = (const float*)d_in[1];
    const float* Wr = (const float*)d_in[2];
    const float* br = (const float*)d_in[3];
    const float* Ws = (const float*)d_in[4];
    const float* bs = (const float*)d_in[5];
    const float* Wo = (const float*)d_in[6];
    const float* bo = (const float*)d_in[7];
    // d_in[8] (area_idx) is arange(N) -> identity permutation; folded out.

    float* gate = (float*)d_ws;                       // 4 floats
    float* Znew = (float*)((char*)d_ws + 256);        // 32*8192 floats (1 MB)

    gate_kernel <<<N_AREAS,      256, 0, stream>>>(Z0, gate);
    fused_kernel<<<N_TOT / 64,   128, 0, stream>>>(x, Z0, Wr, br, Ws, bs, gate, Znew);
    out_kernel  <<<B_SZ * OUTD,  256, 0, stream>>>(Znew, Wo, bo, (float*)d_out);
}